// RGCN_30709016166746
// MI455X (gfx1250) — compile-verified
//
#include <hip/hip_runtime.h>
#include <math.h>

typedef __attribute__((ext_vector_type(2))) float v2f;
typedef __attribute__((ext_vector_type(8))) float v8f;

#define N_NODES 100000
#define N_EDGES 1600000
#define R_REL   4
#define NB_B    30
#define F_INN   86
#define KPAD1   88    /* F_IN padded to multiple of 4 */
#define HDIM    128
#define NCLS    18
#define NGRAPH  512
#define WCOLS   640   /* R*H + H (root) */

// ---------------------------------------------------------------------------
// Zero-pad x [N x 86] -> Xp [N x 88] so the GEMM needs no K guards.
// ---------------------------------------------------------------------------
__global__ void rgcn_pad_x(const float* __restrict__ x, float* __restrict__ xp) {
  size_t idx = (size_t)blockIdx.x * blockDim.x + threadIdx.x;
  if (idx >= (size_t)N_NODES * KPAD1) return;
  size_t n = idx / KPAD1;
  int k = (int)(idx % KPAD1);
  xp[idx] = (k < F_INN) ? x[n * F_INN + k] : 0.0f;
}

// ---------------------------------------------------------------------------
// Build concatenated weight in k-pair-packed layout:
//   logical W[kpad][640]:  cols [r*128, r*128+128) = sum_b comp[r,b]*basis[b,k,o]
//                          cols [512, 640)         = root[k,o]
//                          rows k >= kin           = 0  (padding)
//   stored as Wp[(k>>1)*640 + c][2] with [k&1] selecting the pair element,
//   so one b64 load fetches the (k, k+1) B-operand pair for a column.
// ---------------------------------------------------------------------------
__global__ void rgcn_build_wcat(const float* __restrict__ basis,
                                const float* __restrict__ comp,
                                const float* __restrict__ root,
                                float* __restrict__ wp, int kin, int kpad) {
  int idx = blockIdx.x * blockDim.x + threadIdx.x;
  if (idx >= kpad * WCOLS) return;
  int k = idx / WCOLS;
  int c = idx % WCOLS;
  float val = 0.0f;
  if (k < kin) {
    if (c < R_REL * HDIM) {
      int r = c / HDIM, o = c % HDIM;
      float acc = 0.0f;
#pragma unroll 6
      for (int b = 0; b < NB_B; ++b)
        acc += comp[r * NB_B + b] * basis[((size_t)b * kin + k) * HDIM + o];
      val = acc;
    } else {
      val = root[(size_t)k * HDIM + (c - R_REL * HDIM)];
    }
  }
  wp[((size_t)(k >> 1) * WCOLS + c) * 2 + (k & 1)] = val;
}

// ---------------------------------------------------------------------------
// Edge degree per (relation, dst) segment.
// ---------------------------------------------------------------------------
__global__ void rgcn_edge_count(const int* __restrict__ etype,
                                const int* __restrict__ dst,
                                int* __restrict__ cnt) {
  int e = blockIdx.x * blockDim.x + threadIdx.x;
  if (e >= N_EDGES) return;
  atomicAdd(&cnt[(size_t)etype[e] * N_NODES + dst[e]], 1);
}

// ---------------------------------------------------------------------------
// WMMA GEMM: Y[M x 640] = X[M x Kpad] * W[Kpad x 640] (k-pair-packed).
// Requirements (met by construction): M % 16 == 0, Kpad % 4 == 0.
// One wave computes a 16x64 slab: 4 fp32 accumulators, A reused 4x.
// Inner loop: 1 b64 A-load + 4 b64 B-loads + 4 v_wmma_f32_16x16x4_f32.
// ---------------------------------------------------------------------------
__global__ void rgcn_gemm_wmma(const float* __restrict__ X,
                               const float* __restrict__ Wp,
                               float* __restrict__ Y,
                               int M, int Kpad) {
  int wid = blockIdx.x * (blockDim.x >> 5) + (threadIdx.x >> 5);
  int rowTiles = M >> 4;
  int colSlabs = WCOLS / 64;                 // 10
  if (wid >= rowTiles * colSlabs) return;    // wave-uniform exit
  int rowTile = wid / colSlabs;
  int colBase = (wid % colSlabs) * 64;

  int lane = threadIdx.x & 31;
  int half = lane >> 4;                      // K sub-pair selector
  int mn   = lane & 15;                      // row (A) / col (B,C)

  // A: (k, k+1) pair for row (rowTile*16+mn); lanes 16-31 start at k=2.
  const v2f* xptr = (const v2f*)(X + (size_t)(rowTile * 16 + mn) * Kpad) + half;
  // B: pair-row (k>>1) advances by 2 per k-step; lanes 16-31 offset one pair-row.
  const v2f* wptr = (const v2f*)Wp + ((size_t)half * WCOLS + colBase + mn);

  v8f acc0 = {}, acc1 = {}, acc2 = {}, acc3 = {};

  int steps = Kpad >> 2;
  for (int s = 0; s < steps; ++s) {
    v2f av = *xptr;
    v2f b0 = wptr[0];
    v2f b1 = wptr[16];
    v2f b2 = wptr[32];
    v2f b3 = wptr[48];
    xptr += 2;                // +4 floats along the row
    wptr += 2 * WCOLS;        // +2 pair-rows (= 4 k values)
    acc0 = __builtin_amdgcn_wmma_f32_16x16x4_f32(false, av, false, b0,
                                                 (short)0, acc0, false, false);
    acc1 = __builtin_amdgcn_wmma_f32_16x16x4_f32(false, av, false, b1,
                                                 (short)0, acc1, false, false);
    acc2 = __builtin_amdgcn_wmma_f32_16x16x4_f32(false, av, false, b2,
                                                 (short)0, acc2, false, false);
    acc3 = __builtin_amdgcn_wmma_f32_16x16x4_f32(false, av, false, b3,
                                                 (short)0, acc3, false, false);
  }

  // C/D layout: VGPR v holds row (half*8 + v), col = mn.
  float* yr = Y + (size_t)(rowTile * 16 + (half << 3)) * WCOLS + colBase + mn;
#pragma unroll
  for (int v = 0; v < 8; ++v) {
    yr[0]  = acc0[v];
    yr[16] = acc1[v];
    yr[32] = acc2[v];
    yr[48] = acc3[v];
    yr += WCOLS;
  }
}

// ---------------------------------------------------------------------------
// Edge scatter: one wave per edge; lane handles 4 of 128 channels.
// agg[dst] += xw[src, etype*128 + ch] / cnt[etype, dst]   (fused mean+rel-sum)
// ---------------------------------------------------------------------------
__global__ void rgcn_edge_scatter(const float* __restrict__ xw,
                                  const int* __restrict__ cnt,
                                  const int* __restrict__ src,
                                  const int* __restrict__ dst,
                                  const int* __restrict__ etype,
                                  float* __restrict__ agg) {
  int e = blockIdx.x * (blockDim.x >> 5) + (threadIdx.x >> 5);
  if (e >= N_EDGES) return;
  int lane = threadIdx.x & 31;
  int r = etype[e];
  int s = src[e];
  int d = dst[e];
  int c = cnt[(size_t)r * N_NODES + d];
  float inv = 1.0f / (float)(c > 1 ? c : 1);
  const float4* p = (const float4*)(xw + (size_t)s * WCOLS + r * HDIM);
  float4 v = p[lane];
  float* ap = agg + (size_t)d * HDIM + (lane << 2);
  unsafeAtomicAdd(ap + 0, v.x * inv);
  unsafeAtomicAdd(ap + 1, v.y * inv);
  unsafeAtomicAdd(ap + 2, v.z * inv);
  unsafeAtomicAdd(ap + 3, v.w * inv);
}

// ---------------------------------------------------------------------------
// h = (agg + root_part + bias), optional ReLU, in place over agg.
// root part lives in cols [512,640) of the concatenated GEMM output.
// ---------------------------------------------------------------------------
__global__ void rgcn_bias_act(float* __restrict__ agg,
                              const float* __restrict__ xw,
                              const float* __restrict__ bias,
                              int do_relu) {
  size_t idx = (size_t)blockIdx.x * blockDim.x + threadIdx.x;
  if (idx >= (size_t)N_NODES * HDIM) return;
  size_t nidx = idx / HDIM;
  int o = (int)(idx % HDIM);
  float v = agg[idx] + xw[nidx * WCOLS + R_REL * HDIM + o] + bias[o];
  if (do_relu) v = fmaxf(v, 0.0f);
  agg[idx] = v;
}

// ---------------------------------------------------------------------------
// global_add_pool: g[batch[n]] += h[n]
// ---------------------------------------------------------------------------
__global__ void rgcn_pool(const float* __restrict__ h,
                          const int* __restrict__ batch,
                          float* __restrict__ g) {
  size_t idx = (size_t)blockIdx.x * blockDim.x + threadIdx.x;
  if (idx >= (size_t)N_NODES * HDIM) return;
  size_t nidx = idx / HDIM;
  int o = (int)(idx % HDIM);
  unsafeAtomicAdd(&g[(size_t)batch[nidx] * HDIM + o], h[idx]);
}

// ---------------------------------------------------------------------------
// Head: relu(g @ fc_w + fc_b) -> log_softmax.  One thread per graph (512).
// ---------------------------------------------------------------------------
__global__ void rgcn_head(const float* __restrict__ g,
                          const float* __restrict__ fcw,
                          const float* __restrict__ fcb,
                          float* __restrict__ out) {
  int gi = blockIdx.x * blockDim.x + threadIdx.x;
  if (gi >= NGRAPH) return;
  float vals[NCLS];
  const float* gr = g + (size_t)gi * HDIM;
#pragma unroll 2
  for (int c = 0; c < NCLS; ++c) {
    float acc = fcb[c];
    for (int h = 0; h < HDIM; ++h) acc += gr[h] * fcw[h * NCLS + c];
    vals[c] = fmaxf(acc, 0.0f);
  }
  float mx = vals[0];
#pragma unroll
  for (int c = 1; c < NCLS; ++c) mx = fmaxf(mx, vals[c]);
  float se = 0.0f;
#pragma unroll
  for (int c = 0; c < NCLS; ++c) se += expf(vals[c] - mx);
  float lse = logf(se) + mx;
#pragma unroll
  for (int c = 0; c < NCLS; ++c) out[(size_t)gi * NCLS + c] = vals[c] - lse;
}

// ---------------------------------------------------------------------------
extern "C" void kernel_launch(void* const* d_in, const int* in_sizes, int n_in,
                              void* d_out, int out_size, void* d_ws, size_t ws_size,
                              hipStream_t stream) {
  const float* x      = (const float*)d_in[0];
  const int*   ei     = (const int*)  d_in[1];   // [2,E]: src row then dst row
  const int*   etype  = (const int*)  d_in[2];
  const int*   batch  = (const int*)  d_in[3];
  const float* basis1 = (const float*)d_in[4];
  const float* comp1  = (const float*)d_in[5];
  const float* root1  = (const float*)d_in[6];
  const float* bias1  = (const float*)d_in[7];
  const float* basis2 = (const float*)d_in[8];
  const float* comp2  = (const float*)d_in[9];
  const float* root2  = (const float*)d_in[10];
  const float* bias2  = (const float*)d_in[11];
  const float* fcw    = (const float*)d_in[12];
  const float* fcb    = (const float*)d_in[13];
  float* out = (float*)d_out;

  const int* src = ei;
  const int* dst = ei + N_EDGES;

  // ---- workspace carve-up (floats) ----
  float* ws   = (float*)d_ws;
  float* Wp1  = ws;                                      // 88*640  = 56320
  float* Wp2  = Wp1 + (size_t)KPAD1 * WCOLS;             // 128*640 = 81920
  int*   cnt  = (int*)(Wp2 + (size_t)HDIM * WCOLS);      // R*N     = 400000
  float* Xp   = (float*)(cnt + (size_t)R_REL * N_NODES); // N*88
  float* xw   = Xp   + (size_t)N_NODES * KPAD1;          // N*640
  float* agg1 = xw   + (size_t)N_NODES * WCOLS;          // N*128
  float* agg2 = agg1 + (size_t)N_NODES * HDIM;           // N*128
  float* gbuf = agg2 + (size_t)N_NODES * HDIM;           // G*128

  hipMemsetAsync(cnt,  0, (size_t)R_REL * N_NODES * sizeof(int),  stream);
  hipMemsetAsync(agg1, 0, (size_t)N_NODES * HDIM * sizeof(float), stream);
  hipMemsetAsync(agg2, 0, (size_t)N_NODES * HDIM * sizeof(float), stream);
  hipMemsetAsync(gbuf, 0, (size_t)NGRAPH * HDIM * sizeof(float),  stream);

  // padded activations + packed weights
  rgcn_pad_x<<<(int)(((size_t)N_NODES * KPAD1 + 255) / 256), 256, 0, stream>>>(x, Xp);
  rgcn_build_wcat<<<(KPAD1 * WCOLS + 255) / 256, 256, 0, stream>>>(
      basis1, comp1, root1, Wp1, F_INN, KPAD1);
  rgcn_build_wcat<<<(HDIM * WCOLS + 255) / 256, 256, 0, stream>>>(
      basis2, comp2, root2, Wp2, HDIM, HDIM);

  // per-(relation,dst) edge counts (shared by both layers)
  rgcn_edge_count<<<(N_EDGES + 255) / 256, 256, 0, stream>>>(etype, dst, cnt);

  int rowTiles = N_NODES / 16;                 // 6250 (exact)
  int waves    = rowTiles * (WCOLS / 64);      // 62500
  int gemmBlks = (waves + 7) / 8;              // 256 thr = 8 waves
  int scatBlks = (N_EDGES + 7) / 8;
  int nhBlks   = (int)(((size_t)N_NODES * HDIM + 255) / 256);

  // ---- layer 1 ----
  rgcn_gemm_wmma<<<gemmBlks, 256, 0, stream>>>(Xp, Wp1, xw, N_NODES, KPAD1);
  rgcn_edge_scatter<<<scatBlks, 256, 0, stream>>>(xw, cnt, src, dst, etype, agg1);
  rgcn_bias_act<<<nhBlks, 256, 0, stream>>>(agg1, xw, bias1, 1);   // ReLU

  // ---- layer 2 (h1 == agg1, stride 128 = Kpad) ----
  rgcn_gemm_wmma<<<gemmBlks, 256, 0, stream>>>(agg1, Wp2, xw, N_NODES, HDIM);
  rgcn_edge_scatter<<<scatBlks, 256, 0, stream>>>(xw, cnt, src, dst, etype, agg2);
  rgcn_bias_act<<<nhBlks, 256, 0, stream>>>(agg2, xw, bias2, 0);   // no ReLU

  // ---- pool + head ----
  rgcn_pool<<<nhBlks, 256, 0, stream>>>(agg2, batch, gbuf);
  rgcn_head<<<(NGRAPH + 255) / 256, 256, 0, stream>>>(gbuf, fcw, fcb, out);
}